// QLayer_87917980549393
// MI455X (gfx1250) — compile-verified
//
#include <hip/hip_runtime.h>
#include <stdint.h>

#define NQ    14
#define NL    16
#define DIM   16384     // 2^14 amplitudes
#define BATCH 512

typedef __attribute__((ext_vector_type(2))) float v2f;
typedef __attribute__((ext_vector_type(8))) float v8f;

__device__ __forceinline__ float2 cmul(float2 a, float2 b) {
    return make_float2(a.x * b.x - a.y * b.y, a.x * b.y + a.y * b.x);
}

// ---------------------------------------------------------------------------
// Compile-time linear algebra over GF(2) for the CNOT ring permutation.
// ring = CNOT(13,0)∘...∘CNOT(0,1) acting on index bits (qubit q <-> bit 13-q).
// Physical slot of virtual index v after l layers: P_l(v), P_l = ring^{-l}.
// img[l][b]  = P_l(e_b)  (XOR-fold columns)
// rowq[q]    = parity mask: bit (13-q) of ring^{16}(p) = parity(rowq[q] & p)
// ---------------------------------------------------------------------------
struct Maps {
    unsigned short img[NL][NQ];
    unsigned short rowq[NQ];
};

constexpr Maps make_maps() {
    Maps m{};
    unsigned r1[NQ] = {};
    for (int b = 0; b < NQ; ++b) r1[b] = 1u << b;
    for (int q = NQ - 1; q >= 0; --q) {          // ring^{-1}: CNOTs reversed
        int pc = 13 - q, pt = 13 - ((q + 1) % NQ);
        for (int b = 0; b < NQ; ++b) {
            unsigned v = r1[b];
            v ^= ((v >> pc) & 1u) << pt;
            r1[b] = v;
        }
    }
    unsigned cur[NQ] = {};
    for (int b = 0; b < NQ; ++b) cur[b] = 1u << b;
    for (int l = 0; l < NL; ++l) {               // P_l = ring^{-l}
        for (int b = 0; b < NQ; ++b) m.img[l][b] = (unsigned short)cur[b];
        unsigned nxt[NQ] = {};
        for (int b = 0; b < NQ; ++b) {
            unsigned acc = 0, src = r1[b];
            for (int i = 0; i < NQ; ++i)
                if ((src >> i) & 1u) acc ^= cur[i];
            nxt[b] = acc;
        }
        for (int b = 0; b < NQ; ++b) cur[b] = nxt[b];
    }
    unsigned f[NQ] = {};
    for (int b = 0; b < NQ; ++b) f[b] = 1u << b;
    for (int rep = 0; rep < NL; ++rep)           // forward ring^16
        for (int q = 0; q < NQ; ++q) {
            int pc = 13 - q, pt = 13 - ((q + 1) % NQ);
            for (int b = 0; b < NQ; ++b) {
                unsigned v = f[b];
                v ^= ((v >> pc) & 1u) << pt;
                f[b] = v;
            }
        }
    for (int q = 0; q < NQ; ++q) {
        int beta = 13 - q;
        unsigned mrow = 0;
        for (int i = 0; i < NQ; ++i) mrow |= ((f[i] >> beta) & 1u) << i;
        m.rowq[q] = (unsigned short)mrow;
    }
    return m;
}

__device__ const Maps MAPS = make_maps();

// Qubit-group bit positions (index bit position of nibble bit j).
// g0:{q0-3} g1:{q4-7} g2:{q8-11} g3:{q12,q13}+identity dummies at q0,q1
constexpr int RPOS[4][4] = {
    {10, 11, 12, 13}, {6, 7, 8, 9}, {2, 3, 4, 5}, {12, 13, 0, 1}};
constexpr int CPOS[4][10] = {
    {0, 1, 2, 3, 4, 5, 6, 7, 8, 9},
    {0, 1, 2, 3, 4, 5, 10, 11, 12, 13},
    {0, 1, 6, 7, 8, 9, 10, 11, 12, 13},
    {2, 3, 4, 5, 6, 7, 8, 9, 10, 11}};

// ---------------------------------------------------------------------------
// Build doubled-real 32x32 gate matrices per layer:
//   Utilde[2i+a][2j+b] from complex U[i][j]:
//     [2i][2j]=Re  [2i][2j+1]=-Im  [2i+1][2j]=Im  [2i+1][2j+1]=Re
// Groups g0..g2 = kron of 4 Rots; g3 = kron(Rot12, Rot13, I, I).
// Layout: Wf[(layer*4+g)*1024 + row*32 + col]   (one stage = 4 KB)
// ---------------------------------------------------------------------------
__global__ void build_mats(const float* __restrict__ params, float* __restrict__ Wf) {
    const int layer = blockIdx.x;
    const int tid = threadIdx.x;  // 256

    __shared__ float2 M[NQ][2][2];
    if (tid < NQ) {
        const float* p = params + layer * (NQ * 3) + tid * 3;
        float phi = p[0], th = p[1], om = p[2];
        float st, ct;  sincosf(th * 0.5f, &st, &ct);
        float a = (phi + om) * 0.5f, bb = (phi - om) * 0.5f;
        float sa, ca, sb, cb;
        sincosf(a, &sa, &ca);
        sincosf(bb, &sb, &cb);
        M[tid][0][0] = make_float2( ca * ct, -sa * ct);
        M[tid][0][1] = make_float2(-cb * st, -sb * st);
        M[tid][1][0] = make_float2( cb * st, -sb * st);
        M[tid][1][1] = make_float2( ca * ct,  sa * ct);
    }
    __syncthreads();

    float* WL = Wf + layer * 4096;
    const int i = tid >> 4, j = tid & 15;
    for (int g = 0; g < 4; ++g) {
        float2 u;
        if (g < 3) {
            u = M[4 * g + 0][(i >> 3) & 1][(j >> 3) & 1];
            u = cmul(u, M[4 * g + 1][(i >> 2) & 1][(j >> 2) & 1]);
            u = cmul(u, M[4 * g + 2][(i >> 1) & 1][(j >> 1) & 1]);
            u = cmul(u, M[4 * g + 3][i & 1][j & 1]);
        } else {
            u = cmul(M[12][(i >> 3) & 1][(j >> 3) & 1],
                     M[13][(i >> 2) & 1][(j >> 2) & 1]);
            if ((i & 3) != (j & 3)) u = make_float2(0.f, 0.f);  // identity part
        }
        float* G = WL + g * 1024;
        G[(2 * i) * 32 + 2 * j + 0]     =  u.x;
        G[(2 * i) * 32 + 2 * j + 1]     = -u.y;
        G[(2 * i + 1) * 32 + 2 * j + 0] =  u.y;
        G[(2 * i + 1) * 32 + 2 * j + 1] =  u.x;
    }
}

// ---------------------------------------------------------------------------
// Main simulator: one block per batch element, state resident in LDS.
// Gate matrices streamed into LDS one stage ahead via async-to-LDS DMA.
// ---------------------------------------------------------------------------
__global__ void __launch_bounds__(256)
qsim(const float* __restrict__ x, const float* __restrict__ Wf, float* __restrict__ out) {
    __shared__ float2 S[DIM];            // 128 KB statevector (re, im)
    __shared__ float  Wst[2][1024];      // 2 x 4 KB double-buffered gate matrix
    __shared__ float  csn[2][NQ];

    const int tid = threadIdx.x;
    const int b = blockIdx.x;

    // --- Angle embedding: amp(i) = prod_q (bit? sin : cos) * (-i)^popcount(i)
    if (tid < NQ) {
        float s, c;
        sincosf(x[b * NQ + tid] * 0.5f, &s, &c);
        csn[0][tid] = c;
        csn[1][tid] = s;
    }
    __syncthreads();
    float cq[NQ], sq[NQ];
#pragma unroll
    for (int q = 0; q < NQ; ++q) { cq[q] = csn[0][q]; sq[q] = csn[1][q]; }

    for (int i = tid; i < DIM; i += 256) {
        float m = 1.0f;
#pragma unroll
        for (int q = 0; q < NQ; ++q)
            m *= ((i >> (13 - q)) & 1) ? sq[q] : cq[q];
        int k = __popc((unsigned)i) & 3;
        float re = (k == 0) ? m : ((k == 2) ? -m : 0.0f);
        float im = (k == 1) ? -m : ((k == 3) ? m : 0.0f);
        S[i] = make_float2(re, im);   // P_0 = identity
    }

    // --- Prime stage 0's matrix into Wst[0] (each thread DMAs 16 bytes) ---
    {
        unsigned loff = (unsigned)(uintptr_t)(&Wst[0][tid * 4]);
        unsigned goff = (unsigned)(tid * 16);
        asm volatile("global_load_async_to_lds_b128 %0, %1, %2"
                     :: "v"(loff), "v"(goff), "s"(Wf) : "memory");
    }
    asm volatile("s_wait_asynccnt 0" ::: "memory");
    __syncthreads();

    const int lane = tid & 31;
    const int wave = tid >> 5;
    const int hh   = lane >> 4;      // lane half
    const int n4   = lane & 15;      // column-within-tile / A-row-within-Mtile
    char* Sb = (char*)S;

    for (int layer = 0; layer < NL; ++layer) {
        // byte-shifted permutation columns for this layer (uniform)
        unsigned PB[NQ];
#pragma unroll
        for (int b2 = 0; b2 < NQ; ++b2) PB[b2] = ((unsigned)MAPS.img[layer][b2]) << 3;

#pragma unroll
        for (int g = 0; g < 4; ++g) {
            const int sidx = layer * 4 + g;
            const float* G = &Wst[sidx & 1][0];

            // Prefetch next stage's 4 KB matrix into the other buffer
            // (wraps harmlessly to stage 0 after the last stage).
            {
                const unsigned nst = (unsigned)((sidx + 1) & 63);
                unsigned loff = (unsigned)(uintptr_t)(&Wst[(sidx + 1) & 1][tid * 4]);
                unsigned goff = nst * 4096u + (unsigned)(tid * 16);
                asm volatile("global_load_async_to_lds_b128 %0, %1, %2"
                             :: "v"(loff), "v"(goff), "s"(Wf) : "memory");
            }

            // A fragments from LDS: Mtile0 rows n4, Mtile1 rows 16+n4
            v2f A0[8], A1[8];
#pragma unroll
            for (int kc = 0; kc < 8; ++kc) {
                A0[kc] = *(const v2f*)(G + n4 * 32 + kc * 4 + 2 * hh);
                A1[kc] = *(const v2f*)(G + (16 + n4) * 32 + kc * 4 + 2 * hh);
            }

            // T tables (byte XOR offsets through P): B rows r = kc*2+hh
            unsigned TB[8];
#pragma unroll
            for (int kc = 0; kc < 8; ++kc) {
                unsigned tv = hh ? PB[RPOS[g][0]] : 0u;
                if (kc & 1) tv ^= PB[RPOS[g][1]];
                if (kc & 2) tv ^= PB[RPOS[g][2]];
                if (kc & 4) tv ^= PB[RPOS[g][3]];
                TB[kc] = tv;
            }
            // D rows m = mt*8 + 4*hh + w
            unsigned TD0[4], TD1[4];
#pragma unroll
            for (int w = 0; w < 4; ++w) {
                unsigned tv = hh ? PB[RPOS[g][2]] : 0u;
                if (w & 1) tv ^= PB[RPOS[g][0]];
                if (w & 2) tv ^= PB[RPOS[g][1]];
                TD0[w] = tv;
                TD1[w] = tv ^ PB[RPOS[g][3]];
            }
            // column fold: per-lane low bits (n4) + loop-invariant wave bits
            unsigned pw = 0;
            if (n4 & 1)   pw ^= PB[CPOS[g][0]];
            if (n4 & 2)   pw ^= PB[CPOS[g][1]];
            if (n4 & 4)   pw ^= PB[CPOS[g][2]];
            if (n4 & 8)   pw ^= PB[CPOS[g][3]];
            if (wave & 1) pw ^= PB[CPOS[g][4]];
            if (wave & 2) pw ^= PB[CPOS[g][5]];
            if (wave & 4) pw ^= PB[CPOS[g][6]];

#pragma unroll 1
            for (int it = 0; it < 8; ++it) {     // 8 column-tiles per wave
                unsigned pc_ = pw;               // only 3 variable fold bits
                if (it & 1) pc_ ^= PB[CPOS[g][7]];
                if (it & 2) pc_ ^= PB[CPOS[g][8]];
                if (it & 4) pc_ ^= PB[CPOS[g][9]];

                // Batch all 8 B fragments into distinct register pairs so the
                // 8 ds_load_b64 issue back-to-back and overlap with the WMMAs.
                float2 sv[8];
#pragma unroll
                for (int kc = 0; kc < 8; ++kc)
                    sv[kc] = *(float2*)(Sb + (pc_ ^ TB[kc]));

                v8f D0 = {0, 0, 0, 0, 0, 0, 0, 0};
                v8f D1 = {0, 0, 0, 0, 0, 0, 0, 0};
#pragma unroll
                for (int kc = 0; kc < 8; ++kc) {
                    v2f B = {sv[kc].x, sv[kc].y};
                    D0 = __builtin_amdgcn_wmma_f32_16x16x4_f32(false, A0[kc], false, B, (short)0, D0, false, false);
                    D1 = __builtin_amdgcn_wmma_f32_16x16x4_f32(false, A1[kc], false, B, (short)0, D1, false, false);
                }
#pragma unroll
                for (int w = 0; w < 4; ++w) {
                    *(float2*)(Sb + (pc_ ^ TD0[w])) = make_float2(D0[2 * w], D0[2 * w + 1]);
                    *(float2*)(Sb + (pc_ ^ TD1[w])) = make_float2(D1[2 * w], D1[2 * w + 1]);
                }
            }
            // retire our async prefetch, then stage barrier (state + Wst visibility)
            asm volatile("s_wait_asynccnt 0" ::: "memory");
            __syncthreads();
        }
        // CNOT ring: folded into next stage's addressing (P_{l+1}) — no data motion
    }

    // --- Measure <Z_q>: sign = parity(rowq[q] & physical_index) ---
    unsigned RQ[NQ];
#pragma unroll
    for (int q = 0; q < NQ; ++q) RQ[q] = MAPS.rowq[q];

    float acc[NQ];
#pragma unroll
    for (int q = 0; q < NQ; ++q) acc[q] = 0.0f;
    for (int i = tid; i < DIM; i += 256) {
        float2 a = S[i];
        float pr = a.x * a.x + a.y * a.y;
#pragma unroll
        for (int q = 0; q < NQ; ++q)
            acc[q] += (__popc((unsigned)i & RQ[q]) & 1) ? -pr : pr;
    }
    __syncthreads();
    float* red = (float*)S;                   // reuse state LDS as scratch
#pragma unroll
    for (int q = 0; q < NQ; ++q) red[q * 256 + tid] = acc[q];
    __syncthreads();
    if (tid < NQ) {
        float s = 0.0f;
        for (int t2 = 0; t2 < 256; ++t2) s += red[tid * 256 + t2];
        out[b * NQ + tid] = s;
    }
}

extern "C" void kernel_launch(void* const* d_in, const int* in_sizes, int n_in,
                              void* d_out, int out_size, void* d_ws, size_t ws_size,
                              hipStream_t stream) {
    const float* x      = (const float*)d_in[0];   // (512, 14)
    const float* params = (const float*)d_in[1];   // (16, 1, 14, 3)
    float* Wf = (float*)d_ws;                      // 64 stages * 4 KB = 256 KB

    build_mats<<<NL, 256, 0, stream>>>(params, Wf);
    qsim<<<BATCH, 256, 0, stream>>>(x, Wf, (float*)d_out);
}